// CTRNN_MD_29248727286350
// MI455X (gfx1250) — compile-verified
//
#include <hip/hip_runtime.h>

// CTRNN-MD scan for MI455X (gfx1250, wave32, WMMA).
// Exact-fp32 path: V_WMMA_F32_16X16X4_F32 so the 50-step scan matches the
// f32 reference numerics (no bf16 drift).
//
// Strategy (roofline: ~60 GFLOP fp32 matrix work vs ~0.48 GB traffic ->
// matrix-compute bound; memory floor ~21 us @ 23.3 TB/s):
//  - Each WG owns a 16-row batch tile for ALL 50 time steps (recurrence mixes
//    H, not B), so the sequential scan never leaves the WGP.
//  - W_h (256x256 f32) lives in LDS for the whole kernel (~260 KB) -- only
//    possible with CDNA5's 320 KB WGP LDS.
//  - h is held in WMMA C-layout registers (for the leaky-ReLU update) and
//    republished to a 16-row LDS tile as the next step's A matrix.

typedef float v2f __attribute__((ext_vector_type(2)));
typedef float v8f __attribute__((ext_vector_type(8)));

#define T_STEPS 50
#define BATCH   8192
#define HID     256
#define NIN     33
#define NIN_PAD 36

// LDS strides chosen for bank-conflict-free WMMA fragment access (64 banks):
//  - 260 % 64 == 4  -> banks = (4*n + k) % 64: distinct within a lane half,
//    and 4*dn ≡ 2 (mod 64) has no solution -> no cross-half conflicts.
//  - 37 coprime to 64 -> conflict-free for the W_in B-fragments.
#define WH_STRIDE 260
#define H_STRIDE  260
#define WIN_STRIDE 37

__global__ __launch_bounds__(256, 1)
void ctrnn_md_kernel(const float* __restrict__ x,       // (T,B,33)
                     const int*   __restrict__ sub_id_p,// scalar
                     const float* __restrict__ gates,   // (15,256)
                     const float* __restrict__ W_in,    // (256,33)
                     const float* __restrict__ b_in,    // (256)
                     const float* __restrict__ W_h,     // (256,256)
                     const float* __restrict__ b_h,     // (256)
                     float* __restrict__ out,           // (T,B,256)
                     float* __restrict__ h_last)        // (B,256)
{
    // B-matrix view:  Bh[k][n]  = W_h[n][k]   stored as Wh_lds[n*260 + k]
    //                 Bin[i][n] = W_in[n][i]  stored as Win_lds[n*37 + i] (i>=33 zero)
    __shared__ float Wh_lds [HID * WH_STRIDE];   // 266,240 B
    __shared__ float Win_lds[HID * WIN_STRIDE];  //  37,888 B
    __shared__ float h_lds  [16  * H_STRIDE];    //  16,640 B  (A-matrix for next step)

    const int tid    = threadIdx.x;
    const int lane   = tid & 31;
    const int wave   = tid >> 5;       // 0..7
    const int half   = lane >> 4;      // 0 or 1 (lane half)
    const int nloc   = lane & 15;
    const int b_base = blockIdx.x * 16;

    // ---- stage weights into LDS (coalesced global reads, linear LDS writes) ----
    for (int idx = tid; idx < HID * HID; idx += 256) {
        const int n = idx >> 8, k = idx & 255;
        Wh_lds[n * WH_STRIDE + k] = W_h[idx];
    }
    for (int idx = tid; idx < HID * NIN_PAD; idx += 256) {
        const int n = idx / NIN_PAD, i = idx - n * NIN_PAD;
        Win_lds[n * WIN_STRIDE + i] = (i < NIN) ? W_in[n * NIN + i] : 0.0f;
    }
    for (int idx = tid; idx < 16 * H_STRIDE; idx += 256) h_lds[idx] = 0.0f;

    const int sub = sub_id_p[0];

    // Per-lane column constants for this wave's two 16-wide N tiles.
    const int n0 = wave * 32 + nloc;
    const int n1 = n0 + 16;
    const float g0  = gates[sub * HID + n0];
    const float g1  = gates[sub * HID + n1];
    const float bh0 = b_h[n0],  bh1 = b_h[n1];
    const float bi0 = b_in[n0], bi1 = b_in[n1];

    // Hidden state kept in C/D register layout:
    //  VGPR r, lane half h -> element (M = r + 8*h, N = n_base + nloc)
    v8f h0acc, h1acc;
#pragma unroll
    for (int r = 0; r < 8; ++r) { h0acc[r] = 0.0f; h1acc[r] = 0.0f; }

    const float* xrow = x + ((size_t)b_base + nloc) * NIN;

    __syncthreads();

    for (int t = 0; t < T_STEPS; ++t) {
        const float* xr = xrow + (size_t)t * BATCH * NIN;

        // Prefetch next step's x rows while we crunch this step.
        if (t + 1 < T_STEPS)
            __builtin_prefetch(xr + (size_t)BATCH * NIN, 0, 1);

        // ---- ext tile: e = x @ W_in^T + b_in  (K padded 33 -> 36) ----
        v8f e0, e1;
#pragma unroll
        for (int r = 0; r < 8; ++r) { e0[r] = bi0; e1[r] = bi1; }
#pragma unroll
        for (int kk = 0; kk < 9; ++kk) {
            const int kh = kk * 4 + 2 * half;           // A/B K index for this lane half
            // Branchless K-tail handling (kh in {32,34} on kk==8): clamp the
            // address (stays in-bounds) and select 0 -> v_cndmask, no exec branch.
            const int i0 = (kh     < NIN) ? kh     : 0;
            const int i1 = (kh + 1 < NIN) ? kh + 1 : 0;
            v2f a;
            a.x = (kh     < NIN) ? xr[i0] : 0.0f;
            a.y = (kh + 1 < NIN) ? xr[i1] : 0.0f;
            v2f p, q;
            p.x = Win_lds[n0 * WIN_STRIDE + kh];
            p.y = Win_lds[n0 * WIN_STRIDE + kh + 1];
            q.x = Win_lds[n1 * WIN_STRIDE + kh];
            q.y = Win_lds[n1 * WIN_STRIDE + kh + 1];
            e0 = __builtin_amdgcn_wmma_f32_16x16x4_f32(false, a, false, p, (short)0, e0, false, false);
            e1 = __builtin_amdgcn_wmma_f32_16x16x4_f32(false, a, false, q, (short)0, e1, false, false);
        }

        // ---- rec tile: rec = h @ W_h^T + b_h  (K = 256, fp32-exact) ----
        // Two independent accumulator chains per tile (even/odd k-step) so the
        // SIMD sees 4 independent WMMA dependency chains per wave.
        v8f r0a, r0b, r1a, r1b;
#pragma unroll
        for (int r = 0; r < 8; ++r) { r0a[r] = bh0; r1a[r] = bh1; r0b[r] = 0.0f; r1b[r] = 0.0f; }
#pragma unroll 8
        for (int kk = 0; kk < 64; kk += 2) {
            {   // even k-step -> chain A
                const int kh = kk * 4 + 2 * half;
                v2f a;                                  // A: h rows (shared across tiles)
                a.x = h_lds[nloc * H_STRIDE + kh];
                a.y = h_lds[nloc * H_STRIDE + kh + 1];
                v2f p, q;                               // B: W_h^T columns
                p.x = Wh_lds[n0 * WH_STRIDE + kh];
                p.y = Wh_lds[n0 * WH_STRIDE + kh + 1];
                q.x = Wh_lds[n1 * WH_STRIDE + kh];
                q.y = Wh_lds[n1 * WH_STRIDE + kh + 1];
                r0a = __builtin_amdgcn_wmma_f32_16x16x4_f32(false, a, false, p, (short)0, r0a, false, false);
                r1a = __builtin_amdgcn_wmma_f32_16x16x4_f32(false, a, false, q, (short)0, r1a, false, false);
            }
            {   // odd k-step -> chain B
                const int kh = (kk + 1) * 4 + 2 * half;
                v2f a;
                a.x = h_lds[nloc * H_STRIDE + kh];
                a.y = h_lds[nloc * H_STRIDE + kh + 1];
                v2f p, q;
                p.x = Wh_lds[n0 * WH_STRIDE + kh];
                p.y = Wh_lds[n0 * WH_STRIDE + kh + 1];
                q.x = Wh_lds[n1 * WH_STRIDE + kh];
                q.y = Wh_lds[n1 * WH_STRIDE + kh + 1];
                r0b = __builtin_amdgcn_wmma_f32_16x16x4_f32(false, a, false, p, (short)0, r0b, false, false);
                r1b = __builtin_amdgcn_wmma_f32_16x16x4_f32(false, a, false, q, (short)0, r1b, false, false);
            }
        }

        // ---- elementwise: h' = relu(0.5*h + 0.5*(e + g*rec)), stream to out ----
        float* orow = out + ((size_t)t * BATCH + b_base) * HID;
#pragma unroll
        for (int r = 0; r < 8; ++r) {
            const float rec0 = r0a[r] + r0b[r];
            const float rec1 = r1a[r] + r1b[r];
            const float p0 = e0[r] + g0 * rec0;
            const float p1 = e1[r] + g1 * rec1;
            h0acc[r] = fmaxf(0.0f, 0.5f * h0acc[r] + 0.5f * p0);
            h1acc[r] = fmaxf(0.0f, 0.5f * h1acc[r] + 0.5f * p1);
            const int M = r + 8 * half;
            orow[(size_t)M * HID + n0] = h0acc[r];
            orow[(size_t)M * HID + n1] = h1acc[r];
        }

        // ---- publish h' as next step's A matrix ----
        __syncthreads();                // all waves done reading h_lds (step t)
#pragma unroll
        for (int r = 0; r < 8; ++r) {
            const int M = r + 8 * half;
            h_lds[M * H_STRIDE + n0] = h0acc[r];
            h_lds[M * H_STRIDE + n1] = h1acc[r];
        }
        __syncthreads();                // h_lds ready for step t+1
    }

    // ---- h_last ----
#pragma unroll
    for (int r = 0; r < 8; ++r) {
        const int M = r + 8 * half;
        h_last[((size_t)b_base + M) * HID + n0] = h0acc[r];
        h_last[((size_t)b_base + M) * HID + n1] = h1acc[r];
    }
}

extern "C" void kernel_launch(void* const* d_in, const int* in_sizes, int n_in,
                              void* d_out, int out_size, void* d_ws, size_t ws_size,
                              hipStream_t stream) {
    (void)in_sizes; (void)n_in; (void)out_size; (void)d_ws; (void)ws_size;
    const float* x     = (const float*)d_in[0];
    const int*   sub   = (const int*)  d_in[1];
    const float* gates = (const float*)d_in[2];
    const float* W_in  = (const float*)d_in[3];
    const float* b_in  = (const float*)d_in[4];
    const float* W_h   = (const float*)d_in[5];
    const float* b_h   = (const float*)d_in[6];
    float* out    = (float*)d_out;
    float* h_last = out + (size_t)T_STEPS * BATCH * HID;

    dim3 grid(BATCH / 16);   // 512 workgroups, each owns 16 batch rows for all T
    dim3 block(256);         // 8 wave32s
    hipLaunchKernelGGL(ctrnn_md_kernel, grid, block, 0, stream,
                       x, sub, gates, W_in, b_in, W_h, b_h, out, h_last);
}